// black_3599182594540
// MI455X (gfx1250) — compile-verified
//
#include <hip/hip_runtime.h>

#define NCLASSES 10
#define KSEL 5
#define EPSF 1e-7f

#define BLOCK 256
#define ROWS_PER_BLOCK 512
#define CHUNKS (ROWS_PER_BLOCK * NCLASSES / 4)   /* 1280 float4 tiles per block */
#define CHUNKS_PER_THREAD (CHUNKS / BLOCK)       /* 5 */

#if defined(__gfx1250__)
typedef int v4i_t __attribute__((ext_vector_type(4)));
typedef __attribute__((address_space(1))) v4i_t* gasp_t;  // global AS pointer
typedef __attribute__((address_space(3))) v4i_t* lasp_t;  // LDS AS pointer
#endif

__device__ __forceinline__ float row_loss(const float* __restrict__ v,
                                          int c, const int* __restrict__ indv)
{
    float m = v[0];
    #pragma unroll
    for (int j = 1; j < NCLASSES; ++j) m = fmaxf(m, v[j]);

    const float e0 = 0.8f * __expf(v[c] - m);
    float ek[KSEL];
    float S = e0;
    #pragma unroll
    for (int k = 0; k < KSEL; ++k) {
        const int col = indv[k] + (indv[k] >= c ? 1 : 0);  // skip label column
        ek[k] = 0.2f * __expf(v[col] - m);
        S += ek[k];
    }
    const float inv = 1.0f / (S + EPSF);
    float l = __logf(e0 * inv + EPSF);
    #pragma unroll
    for (int k = 0; k < KSEL; ++k) l += __logf(1.0f - ek[k] * inv + EPSF);
    return -l;
}

__global__ __launch_bounds__(BLOCK)
void ce_partial_kernel(const float* __restrict__ yHat,
                       const int* __restrict__ y,
                       const int* __restrict__ ind,
                       double* __restrict__ partial,
                       int B)
{
    __shared__ float tile[ROWS_PER_BLOCK * NCLASSES];  // 20 KB
    __shared__ float red[BLOCK];

    const int t = threadIdx.x;
    const long long rowBase    = (long long)blockIdx.x * ROWS_PER_BLOCK;
    const long long chunkBase  = rowBase * NCLASSES / 4;
    const long long totalChunk = ((long long)B * NCLASSES) / 4;
    const bool fullBlock       = (chunkBase + CHUNKS) <= totalChunk;  // wave-uniform

    const float4* g4 = (const float4*)yHat + chunkBase;
    float4*       l4 = (float4*)tile;

#if defined(__gfx1250__) && __has_builtin(__builtin_amdgcn_global_load_async_to_lds_b128) && __has_builtin(__builtin_amdgcn_s_wait_asynccnt)
    // CDNA5 async path: 16B/lane global -> LDS DMA, tracked by ASYNCcnt.
    if (fullBlock) {
        // scalar-uniform fast path: 5 back-to-back async loads, no exec masking
        #pragma unroll
        for (int c = 0; c < CHUNKS_PER_THREAD; ++c) {
            const int k = t + c * BLOCK;
            __builtin_amdgcn_global_load_async_to_lds_b128(
                (gasp_t)(g4 + k), (lasp_t)(l4 + k), /*offset=*/0, /*cpol=*/0);
        }
    } else {
        #pragma unroll
        for (int c = 0; c < CHUNKS_PER_THREAD; ++c) {
            const int k = t + c * BLOCK;
            if (chunkBase + k < totalChunk) {
                __builtin_amdgcn_global_load_async_to_lds_b128(
                    (gasp_t)(g4 + k), (lasp_t)(l4 + k), 0, 0);
            }
        }
    }
    __builtin_amdgcn_s_wait_asynccnt(0);
#else
    #pragma unroll
    for (int c = 0; c < CHUNKS_PER_THREAD; ++c) {
        const int k = t + c * BLOCK;
        if (chunkBase + k < totalChunk) l4[k] = g4[k];
    }
#endif
    __syncthreads();

    int indv[KSEL];
    #pragma unroll
    for (int k = 0; k < KSEL; ++k) indv[k] = ind[k];

    float loss_sum = 0.0f;
    if (fullBlock) {
        #pragma unroll
        for (int r = 0; r < ROWS_PER_BLOCK / BLOCK; ++r) {
            const int lrow = t + r * BLOCK;
            // stride-10-dword LDS reads: 32 distinct banks per wave32, conflict-free
            loss_sum += row_loss(tile + lrow * NCLASSES, y[rowBase + lrow], indv);
        }
    } else {
        #pragma unroll
        for (int r = 0; r < ROWS_PER_BLOCK / BLOCK; ++r) {
            const int lrow = t + r * BLOCK;
            const long long row = rowBase + lrow;
            if (row < (long long)B)
                loss_sum += row_loss(tile + lrow * NCLASSES, y[row], indv);
        }
    }

    red[t] = loss_sum;
    __syncthreads();
    for (int s = BLOCK / 2; s > 0; s >>= 1) {
        if (t < s) red[t] += red[t + s];
        __syncthreads();
    }
    if (t == 0) partial[blockIdx.x] = (double)red[0];
}

__global__ __launch_bounds__(BLOCK)
void ce_finalize_kernel(const double* __restrict__ partial,
                        float* __restrict__ out,
                        int nblocks, long long B)
{
    __shared__ double sred[BLOCK];
    const int t = threadIdx.x;
    double s = 0.0;
    for (int i = t; i < nblocks; i += BLOCK) s += partial[i];  // fixed order: deterministic
    sred[t] = s;
    __syncthreads();
    for (int st = BLOCK / 2; st > 0; st >>= 1) {
        if (t < st) sred[t] += sred[t + st];
        __syncthreads();
    }
    if (t == 0) out[0] = (float)(sred[0] / (double)B);
}

extern "C" void kernel_launch(void* const* d_in, const int* in_sizes, int n_in,
                              void* d_out, int out_size, void* d_ws, size_t ws_size,
                              hipStream_t stream)
{
    const float* yHat = (const float*)d_in[0];
    const int*   y    = (const int*)d_in[1];
    const int*   ind  = (const int*)d_in[2];

    const int B = in_sizes[0] / NCLASSES;
    const int nblocks = (B + ROWS_PER_BLOCK - 1) / ROWS_PER_BLOCK;  // 8192 for B=4M

    double* partial = (double*)d_ws;  // nblocks * 8 bytes = 64 KB of scratch

    ce_partial_kernel<<<nblocks, BLOCK, 0, stream>>>(yHat, y, ind, partial, B);
    ce_finalize_kernel<<<1, BLOCK, 0, stream>>>(partial, (float*)d_out,
                                                nblocks, (long long)B);
}